// UniversalBiphasicAxonMapLayer_54142357734086
// MI455X (gfx1250) — compile-verified
//
#include <hip/hip_runtime.h>

typedef __attribute__((ext_vector_type(2))) float v2f;
typedef __attribute__((ext_vector_type(8))) float v8f;

#define B_    2
#define E_    225
#define EPAD  240
#define P_    2401
#define S_    150
#define SPAD  160
#define NT_E  (EPAD / 16)   // 15 electrode tiles
#define NT_S  (SPAD / 16)   // 10 segment tiles

// ---------------------------------------------------------------------------
// Kernel 1: per-(b,e) coefficient precompute.  ws layout (floats):
//   lo  [B*EPAD*2]  : (2*ex*c1, 2*ey*c1)   -> B-matrix K=0,1
//   hi  [B*EPAD*2]  : (-c1, c2)            -> B-matrix K=2,3
//   bias[B*EPAD]    : -(ex^2+ey^2)*c1      -> C accumulator init
//   fb  [B*EPAD]    : F_bright             -> post-exp weight
// ---------------------------------------------------------------------------
__global__ void axonmap_precompute(const float* __restrict__ stimuli,
                                   const float* __restrict__ phis,
                                   const float* __restrict__ elec_x,
                                   const float* __restrict__ elec_y,
                                   float* __restrict__ ws) {
    int idx = blockIdx.x * blockDim.x + threadIdx.x;
    if (idx >= B_ * EPAD) return;
    int b = idx / EPAD;
    int e = idx - b * EPAD;

    float* lo   = ws;
    float* hi   = ws + B_ * EPAD * 2;
    float* bias = ws + B_ * EPAD * 4;
    float* fb   = ws + B_ * EPAD * 5;

    float l0 = 0.f, l1 = 0.f, h0 = 0.f, h1 = 0.f, bi = 0.f, fbr = 0.f;
    if (e < E_) {
        float freq = stimuli[(b * E_ + e) * 3 + 0];
        float amp  = stimuli[(b * E_ + e) * 3 + 1];
        float pdur = stimuli[(b * E_ + e) * 3 + 2];
        const float* ph = phis + b * 12;
        float rho = ph[0], axl = ph[1];
        float a0 = ph[2], a1 = ph[3], a2 = ph[4], a3 = ph[5];
        float a5 = ph[7], a6 = ph[8], a7 = ph[9], a8 = ph[10], a9 = ph[11];

        float scaled  = (a1 + a0 * pdur) * amp;
        float Fbright = (scaled > 0.25f) ? (a2 * scaled + a3 * freq) : 0.0f;
        float Fsize   = fmaxf(a5 * scaled + a6, 100.0f / (rho * rho));
        float Fstreak = fmaxf(a9 - a7 * powf(pdur, a8), 100.0f / (axl * axl));
        float c1 = 1.0f / (2.0f * rho * rho * Fsize);
        float c2 = 1.0f / (axl * axl * Fstreak);
        float ex = elec_x[e], ey = elec_y[e];

        l0 = 2.0f * ex * c1;
        l1 = 2.0f * ey * c1;
        h0 = -c1;
        h1 = c2;
        bi = -(ex * ex + ey * ey) * c1;
        fbr = Fbright;
    }
    lo[idx * 2 + 0] = l0;
    lo[idx * 2 + 1] = l1;
    hi[idx * 2 + 0] = h0;
    hi[idx * 2 + 1] = h1;
    bias[idx] = bi;
    fb[idx]   = fbr;
}

// ---------------------------------------------------------------------------
// Kernel 2: one wave32 per (b,p).  exponent tile via V_WMMA_F32_16X16X4_F32,
// then v_exp_f32, F_bright-weighted accumulation over electrode tiles,
// cross-lane column sum, max over segments, threshold.
// ---------------------------------------------------------------------------
__global__ __launch_bounds__(128) void axonmap_wmma(
        const float* __restrict__ axon,     // (P,S,3): x, y, sens
        const float* __restrict__ ws,
        const float* __restrict__ thresh,
        float* __restrict__ out) {
    const int lane   = threadIdx.x & 31;
    const int n      = lane & 15;          // column / row-in-tile index
    const bool hiK   = lane >= 16;         // lane half selects K pair

    int wid = blockIdx.x * 4 + (threadIdx.x >> 5);
    const bool valid = (wid < B_ * P_);
    if (!valid) wid = B_ * P_ - 1;         // uniform clamp: keep EXEC full for WMMA
    const int b = wid / P_;
    const int p = wid - b * P_;

    const float* lo   = ws;
    const float* hi   = ws + B_ * EPAD * 2;
    const float* bias = ws + B_ * EPAD * 4;
    const float* fbp  = ws + B_ * EPAD * 5;

    const float LOG2E = 1.4426950408889634f;
    float best = 0.0f;

    for (int t = 0; t < NT_S; ++t) {
        // --- A tile: 16 segment-rows x K=4 [x, y, x^2+y^2, sens] ---
        int s = t * 16 + n;
        float x = 0.0f, y = 0.0f, sens = -1e30f;   // pad rows -> exp -> 0
        if (s < S_) {
            const float* row = axon + (p * S_ + s) * 3;
            x = row[0]; y = row[1]; sens = row[2];
        }
        v2f a;
        if (!hiK) { a.x = x;             a.y = y;    }   // K=0,1
        else      { a.x = x * x + y * y; a.y = sens; }   // K=2,3

        v8f acc = {0.f, 0.f, 0.f, 0.f, 0.f, 0.f, 0.f, 0.f};

        for (int et = 0; et < NT_E; ++et) {
            int e = b * EPAD + et * 16 + n;
            // --- B tile: K=4 x 16 electrode-cols ---
            const float* src = hiK ? hi : lo;
            v2f bm;
            bm.x = src[e * 2 + 0];
            bm.y = src[e * 2 + 1];
            float bv = bias[e];
            float fv = fbp[e];
            v8f c = {bv, bv, bv, bv, bv, bv, bv, bv};   // per-column bias

            v8f d = __builtin_amdgcn_wmma_f32_16x16x4_f32(
                false, a, false, bm, (short)0, c, false, false);

#pragma unroll
            for (int i = 0; i < 8; ++i)
                acc[i] += fv * __builtin_amdgcn_exp2f(d[i] * LOG2E);
        }

        // sum over 16 electrode columns (lanes within each half), max over rows
#pragma unroll
        for (int i = 0; i < 8; ++i) {
            float v = acc[i];
            v += __shfl_xor(v, 1, 32);
            v += __shfl_xor(v, 2, 32);
            v += __shfl_xor(v, 4, 32);
            v += __shfl_xor(v, 8, 32);
            best = fmaxf(best, v);
        }
    }
    // merge the two halves (rows 0-7 vs 8-15 of each tile)
    best = fmaxf(best, __shfl_xor(best, 16, 32));

    if (valid && lane == 0) {
        float th = thresh[0];
        out[wid] = (best > th) ? best : 0.0f;
    }
}

// ---------------------------------------------------------------------------
extern "C" void kernel_launch(void* const* d_in, const int* in_sizes, int n_in,
                              void* d_out, int out_size, void* d_ws, size_t ws_size,
                              hipStream_t stream) {
    const float* stimuli = (const float*)d_in[0];   // (B,E,3)
    const float* phis    = (const float*)d_in[1];   // (B,12)
    const float* axon    = (const float*)d_in[2];   // (P,S,3)
    const float* elec_x  = (const float*)d_in[3];   // (E,)
    const float* elec_y  = (const float*)d_in[4];   // (E,)
    const float* thresh  = (const float*)d_in[5];   // (1,)
    float* out = (float*)d_out;                     // (B*P,)
    float* ws  = (float*)d_ws;                      // 2880 floats used

    axonmap_precompute<<<(B_ * EPAD + 255) / 256, 256, 0, stream>>>(
        stimuli, phis, elec_x, elec_y, ws);

    int nwaves = B_ * P_;                // 4802
    int blocks = (nwaves + 3) / 4;       // 4 waves / 128-thread block
    axonmap_wmma<<<blocks, 128, 0, stream>>>(axon, ws, thresh, out);
}